// DecoderGRUWeighted_14027363189411
// MI455X (gfx1250) — compile-verified
//
#include <hip/hip_runtime.h>
#include <math.h>

typedef float v2f __attribute__((ext_vector_type(2)));
typedef float v8f __attribute__((ext_vector_type(8)));

#define H 1024
#define V 50257
#define L 512
#define NB 128   // stat-reduction blocks

// ---------------------------------------------------------------------------
// Generic WMMA GEMV:  y[r] = act( dot(W[r, 0:KD], x) + bias[r] ),  W row-major.
// One wave per 16-row tile. A-side = 16 weight rows (lane m<16 loads
// W[row][k0..k0+1], lane m+16 loads W[row][k0+2..k0+3] -> ISA 32-bit A layout).
// B-side = x broadcast across all 16 columns (lanes<16 load x[k0..k0+1],
// lanes>=16 load x[k0+2..k0+3]). Result column N=0: lane 0 holds M=0..7,
// lane 16 holds M=8..15 in the 8 C/D VGPRs.
// KD is compile-time, multiple of 8. NT=1 streams W with non-temporal hint
// (used for the 206MB Wo matrix, which exceeds the 192MB L2).
// ---------------------------------------------------------------------------
template <int RELU, int NT, int KD>
__global__ __launch_bounds__(256) void gemv_wmma(
    const float* __restrict__ W, const float* __restrict__ x,
    const float* __restrict__ bias, float* __restrict__ y, int R) {
  const int lane = threadIdx.x & 31;
  const int wave = blockIdx.x * (blockDim.x >> 5) + (threadIdx.x >> 5);
  const int v0 = wave << 4;
  if (v0 >= R) return;  // uniform whole-wave exit (EXEC stays all-ones for WMMA)

  int row = v0 + (lane & 15);
  if (row >= R) row = R - 1;            // clamp (dup rows are never stored)
  const int s2 = (lane >> 4) << 1;      // 0 for lanes 0-15, 2 for lanes 16-31

  const float* wp = W + (size_t)row * (size_t)KD + s2;
  const float* xp = x + s2;

  v8f c0 = {0.f, 0.f, 0.f, 0.f, 0.f, 0.f, 0.f, 0.f};
  v8f c1 = {0.f, 0.f, 0.f, 0.f, 0.f, 0.f, 0.f, 0.f};

#pragma unroll 8
  for (int k0 = 0; k0 < KD; k0 += 8) {
    v2f a0, a1;
    if (NT) {
      a0 = __builtin_nontemporal_load((const v2f*)(wp + k0));
      a1 = __builtin_nontemporal_load((const v2f*)(wp + k0 + 4));
    } else {
      a0 = *(const v2f*)(wp + k0);
      a1 = *(const v2f*)(wp + k0 + 4);
    }
    v2f b0 = *(const v2f*)(xp + k0);
    v2f b1 = *(const v2f*)(xp + k0 + 4);
    c0 = __builtin_amdgcn_wmma_f32_16x16x4_f32(false, a0, false, b0, (short)0,
                                               c0, false, false);
    c1 = __builtin_amdgcn_wmma_f32_16x16x4_f32(false, a1, false, b1, (short)0,
                                               c1, false, false);
  }
  v8f c = c0 + c1;

  if ((lane & 15) == 0) {  // lane 0 -> M=0..7, lane 16 -> M=8..15
    const int base = v0 + ((lane >> 4) << 3);
#pragma unroll
    for (int r = 0; r < 8; ++r) {
      const int m = base + r;
      if (m < R) {
        float v = c[r] + bias[m];
        if (RELU) v = fmaxf(v, 0.f);
        y[m] = v;
      }
    }
  }
}

// ---------------------------------------------------------------------------
__global__ __launch_bounds__(256) void build_cat1(const int* __restrict__ x,
                                                  const float* __restrict__ emb,
                                                  const float* __restrict__ hid,
                                                  float* __restrict__ cat1) {
  const int i = blockIdx.x * blockDim.x + threadIdx.x;  // 0..2047
  if (i < H)
    cat1[i] = emb[(size_t)x[0] * H + i];
  else if (i < 2 * H)
    cat1[i] = hid[i - H];
}

__global__ __launch_bounds__(512) void softmax_L(const float* __restrict__ s,
                                                 float* __restrict__ w_out,
                                                 float* __restrict__ w_ws) {
  __shared__ float red[L];
  const int t = threadIdx.x;
  const float v = s[t];
  red[t] = v;
  __syncthreads();
  for (int k = L / 2; k > 0; k >>= 1) {
    if (t < k) red[t] = fmaxf(red[t], red[t + k]);
    __syncthreads();
  }
  const float m = red[0];
  __syncthreads();
  const float e = expf(v - m);
  red[t] = e;
  __syncthreads();
  for (int k = L / 2; k > 0; k >>= 1) {
    if (t < k) red[t] += red[t + k];
    __syncthreads();
  }
  const float w = e / red[0];
  w_out[t] = w;
  w_ws[t] = w;
}

// wctx[h] = sum_l w[l]*enc[l][h]; also assembles cat2 = [embed, wctx]
__global__ __launch_bounds__(256) void attn_ctx(const float* __restrict__ w,
                                                const float* __restrict__ enc,
                                                const float* __restrict__ cat1,
                                                float* __restrict__ cat2) {
  const int h = blockIdx.x * blockDim.x + threadIdx.x;  // 0..1023
  float acc = 0.f;
#pragma unroll 4
  for (int l = 0; l < L; ++l) acc = fmaf(w[l], enc[(size_t)l * H + h], acc);
  cat2[h] = cat1[h];
  cat2[H + h] = acc;
}

__global__ __launch_bounds__(256) void gru_combine(
    const float* __restrict__ gi, const float* __restrict__ gh,
    const float* __restrict__ hid, float* __restrict__ h_ws,
    float* __restrict__ h_out) {
  const int i = blockIdx.x * blockDim.x + threadIdx.x;  // 0..1023
  const float r = 1.f / (1.f + expf(-(gi[i] + gh[i])));
  const float z = 1.f / (1.f + expf(-(gi[H + i] + gh[H + i])));
  const float n = tanhf(gi[2 * H + i] + r * gh[2 * H + i]);
  const float hn = (1.f - z) * n + z * hid[i];
  h_ws[i] = hn;
  h_out[i] = hn;
}

// per-block online (max, sum-exp) over the logits
__global__ __launch_bounds__(256) void logits_stats(
    const float* __restrict__ logits, float* __restrict__ stats) {
  __shared__ float sm[256], ss[256];
  const int t = threadIdx.x;
  float m = -INFINITY, s = 0.f;
  for (int i = blockIdx.x * 256 + t; i < V; i += NB * 256) {
    const float v = logits[i];
    if (v > m) {
      s = s * expf(m - v) + 1.f;
      m = v;
    } else {
      s += expf(v - m);
    }
  }
  sm[t] = m;
  ss[t] = s;
  __syncthreads();
  for (int k = 128; k > 0; k >>= 1) {
    if (t < k) {
      const float m2 = sm[t + k], s2 = ss[t + k];
      const float M = fmaxf(sm[t], m2);
      ss[t] = ss[t] * expf(sm[t] - M) + s2 * expf(m2 - M);
      sm[t] = M;
    }
    __syncthreads();
  }
  if (t == 0) {
    stats[2 * blockIdx.x] = sm[0];
    stats[2 * blockIdx.x + 1] = ss[0];
  }
}

__global__ __launch_bounds__(128) void merge_stats(
    const float* __restrict__ stats, float* __restrict__ msum) {
  __shared__ float sm[NB], ss[NB];
  const int t = threadIdx.x;
  sm[t] = stats[2 * t];
  ss[t] = stats[2 * t + 1];
  __syncthreads();
  for (int k = NB / 2; k > 0; k >>= 1) {
    if (t < k) {
      const float m2 = sm[t + k], s2 = ss[t + k];
      const float M = fmaxf(sm[t], m2);
      ss[t] = ss[t] * expf(sm[t] - M) + s2 * expf(m2 - M);
      sm[t] = M;
    }
    __syncthreads();
  }
  if (t == 0) {
    msum[0] = sm[0];
    msum[1] = logf(ss[0]);
  }
}

__global__ __launch_bounds__(256) void log_softmax_out(
    const float* __restrict__ logits, const float* __restrict__ msum,
    float* __restrict__ out) {
  const int i = blockIdx.x * 256 + threadIdx.x;
  if (i < V) out[i] = logits[i] - msum[0] - msum[1];
}

// ---------------------------------------------------------------------------
extern "C" void kernel_launch(void* const* d_in, const int* in_sizes, int n_in,
                              void* d_out, int out_size, void* d_ws,
                              size_t ws_size, hipStream_t stream) {
  const int* x = (const int*)d_in[0];
  const float* hidden = (const float*)d_in[1];
  const float* enc = (const float*)d_in[2];
  const float* emb = (const float*)d_in[3];
  const float* Ww = (const float*)d_in[4];
  const float* bw = (const float*)d_in[5];
  const float* Wc = (const float*)d_in[6];
  const float* bc = (const float*)d_in[7];
  const float* Wih = (const float*)d_in[8];
  const float* Whh = (const float*)d_in[9];
  const float* bih = (const float*)d_in[10];
  const float* bhh = (const float*)d_in[11];
  const float* Wo = (const float*)d_in[12];
  const float* bo = (const float*)d_in[13];

  float* out_logits = (float*)d_out;       // [V]  log_softmax
  float* out_h = out_logits + V;           // [H]  h_new
  float* out_w = out_h + H;                // [L]  attention weights

  float* ws = (float*)d_ws;
  float* cat1 = ws;                        // 2H
  float* scores = cat1 + 2 * H;            // L
  float* wgt = scores + L;                 // L
  float* cat2 = wgt + L;                   // 2H
  float* outv = cat2 + 2 * H;              // H
  float* gi = outv + H;                    // 3H
  float* gh = gi + 3 * H;                  // 3H
  float* hnew = gh + 3 * H;                // H
  float* logits = hnew + H;                // V (padded)
  float* stats = logits + ((V + 63) & ~63);  // 2*NB
  float* msum = stats + 2 * NB;            // 2

  build_cat1<<<8, 256, 0, stream>>>(x, emb, hidden, cat1);
  // scores = Ww(512x2048) @ cat1 + bw : 32 tiles -> 4 blocks
  gemv_wmma<0, 0, 2 * H><<<4, 256, 0, stream>>>(Ww, cat1, bw, scores, L);
  softmax_L<<<1, 512, 0, stream>>>(scores, out_w, wgt);
  attn_ctx<<<4, 256, 0, stream>>>(wgt, enc, cat1, cat2);
  // out = relu(Wc(1024x2048) @ cat2 + bc) : 64 tiles -> 8 blocks
  gemv_wmma<1, 0, 2 * H><<<8, 256, 0, stream>>>(Wc, cat2, bc, outv, H);
  // gi = Wih(3072x1024) @ out + bih ; gh = Whh @ hidden + bhh : 192 tiles each
  gemv_wmma<0, 0, H><<<24, 256, 0, stream>>>(Wih, outv, bih, gi, 3 * H);
  gemv_wmma<0, 0, H><<<24, 256, 0, stream>>>(Whh, hidden, bhh, gh, 3 * H);
  gru_combine<<<4, 256, 0, stream>>>(gi, gh, hidden, hnew, out_h);
  // logits = Wo(50257x1024) @ h_new + bo : 3142 tiles -> 393 blocks
  // NT=1: Wo (206MB) exceeds L2 (192MB) -> stream non-temporally, keep the
  // small weights (Ww/Wc/Wih/Whh, 36MB) L2-resident across graph replays.
  gemv_wmma<0, 1, H><<<393, 256, 0, stream>>>(Wo, hnew, bo, logits, V);
  logits_stats<<<NB, 256, 0, stream>>>(logits, stats);
  merge_stats<<<1, NB, 0, stream>>>(stats, msum);
  log_softmax_out<<<(V + 255) / 256, 256, 0, stream>>>(logits, msum,
                                                       out_logits);
}